// OutputLayer_15719580304442
// MI455X (gfx1250) — compile-verified
//
#include <hip/hip_runtime.h>
#include <math.h>

#define SEQ 16384
#define DDIM 1024

typedef __attribute__((ext_vector_type(2))) float v2f;
typedef __attribute__((ext_vector_type(8))) float v8f;

// Phase 1: p[head][s] = dot(wp_head, G_head[s,:]) via V_WMMA_F32_16X16X4_F32.
// One wave (32 lanes) handles a 16-row seq tile; 8 waves per block.
__global__ __launch_bounds__(256) void matvec_wmma_kernel(
    const float* __restrict__ G1, const float* __restrict__ G2,
    const float* __restrict__ wp1, const float* __restrict__ wp2,
    float* __restrict__ out)
{
    const int head = blockIdx.y;
    const float* __restrict__ G  = head ? G2  : G1;
    const float* __restrict__ wp = head ? wp2 : wp1;

    const int lane = threadIdx.x & 31;
    const int wave = threadIdx.x >> 5;
    const int s0   = (blockIdx.x * 8 + wave) * 16;

    // B-matrix (4x16 f32) lane mapping: N = lane&15 (seq), K pair = 2*(lane>>4)
    const int n    = lane & 15;
    const int doff = (lane >> 4) * 2;

    const float* __restrict__ grow = G + (size_t)(s0 + n) * DDIM + doff;
    const float* __restrict__ wrow = wp + doff;

    // A-matrix (16x4 f32): only row M=0 is live (lanes 0 and 16 carry wp).
    const bool isRow0 = (n == 0);

    v8f c = {};  // 16x16 f32 accumulator, we only consume M=0 (VGPR 0, lanes 0-15)

#pragma unroll 8
    for (int d0 = 0; d0 < DDIM; d0 += 4) {
        v2f b = *(const v2f*)(grow + d0);     // G[s0+n, d0+doff .. +1]
        v2f w = *(const v2f*)(wrow + d0);     // wp[d0+doff .. +1]
        v2f a;
        a.x = isRow0 ? w.x : 0.0f;
        a.y = isRow0 ? w.y : 0.0f;
        c = __builtin_amdgcn_wmma_f32_16x16x4_f32(
                /*neg_a=*/false, a, /*neg_b=*/false, b,
                /*c_mod=*/(short)0, c, /*reuse_a=*/false, /*reuse_b=*/false);
    }

    // C layout: VGPR0, lanes 0-15 hold M=0, N=lane.
    if (lane < 16) {
        out[head * SEQ + s0 + lane] = c[0];
    }
}

// Phase 2: in-place softmax over 16384 logits, one block per head.
__global__ __launch_bounds__(1024) void softmax_kernel(float* __restrict__ p)
{
    __shared__ float red[1024];
    float* __restrict__ x = p + (size_t)blockIdx.x * SEQ;
    const int tid = threadIdx.x;

    float v[16];
    float m = -INFINITY;
#pragma unroll
    for (int i = 0; i < 16; ++i) {
        v[i] = x[tid + i * 1024];
        m = fmaxf(m, v[i]);
    }
    red[tid] = m;
    __syncthreads();
    for (int s = 512; s > 0; s >>= 1) {
        if (tid < s) red[tid] = fmaxf(red[tid], red[tid + s]);
        __syncthreads();
    }
    const float M = red[0];
    __syncthreads();

    float sum = 0.0f;
#pragma unroll
    for (int i = 0; i < 16; ++i) {
        v[i] = __expf(v[i] - M);
        sum += v[i];
    }
    red[tid] = sum;
    __syncthreads();
    for (int s = 512; s > 0; s >>= 1) {
        if (tid < s) red[tid] += red[tid + s];
        __syncthreads();
    }
    const float inv = 1.0f / red[0];
#pragma unroll
    for (int i = 0; i < 16; ++i) {
        x[tid + i * 1024] = v[i] * inv;
    }
}

extern "C" void kernel_launch(void* const* d_in, const int* in_sizes, int n_in,
                              void* d_out, int out_size, void* d_ws, size_t ws_size,
                              hipStream_t stream)
{
    (void)in_sizes; (void)n_in; (void)out_size; (void)d_ws; (void)ws_size;

    const float* G1  = (const float*)d_in[0];
    const float* G2  = (const float*)d_in[1];
    const float* wp1 = (const float*)d_in[2];
    const float* wp2 = (const float*)d_in[3];
    float* out = (float*)d_out;

    // 2048 tiles of 16 rows; 8 waves (tiles) per 256-thread block; y = head.
    dim3 grid(SEQ / (16 * 8), 2, 1);
    matvec_wmma_kernel<<<grid, dim3(256, 1, 1), 0, stream>>>(G1, G2, wp1, wp2, out);

    softmax_kernel<<<dim3(2, 1, 1), dim3(1024, 1, 1), 0, stream>>>(out);
}